// GCN_39848706573686
// MI455X (gfx1250) — compile-verified
//
#include <hip/hip_runtime.h>
#include <hip/hip_bf16.h>

typedef float v2f __attribute__((ext_vector_type(2)));
typedef float v8f __attribute__((ext_vector_type(8)));

#define F_DIM 128
#define H_DIM 16

// ---------------------------------------------------------------- init
__global__ void gcn_init(float* __restrict__ deg, float* __restrict__ agg1,
                         float* __restrict__ out, int n) {
  int i = blockIdx.x * blockDim.x + threadIdx.x;
  if (i < n * H_DIM) agg1[i] = 0.0f;
  if (i < n) { deg[i] = 1.0f; out[i] = 0.0f; }   // deg starts at 1 (self-loop)
}

// ---------------------------------------------------------------- degree
__global__ void gcn_degree(const int* __restrict__ dst, float* __restrict__ deg, int E) {
  int e = blockIdx.x * blockDim.x + threadIdx.x;
  if (e < E) unsafeAtomicAdd(&deg[dst[e]], 1.0f);
}

// ---------------------------------------------------------------- rsqrt
__global__ void gcn_dis(const float* __restrict__ deg, float* __restrict__ dis, int n) {
  int i = blockIdx.x * blockDim.x + threadIdx.x;
  if (i < n) dis[i] = rsqrtf(deg[i]);            // deg >= 1 always (self-loops)
}

// ---------------------------------------------------------------- h = x @ W1 via V_WMMA_F32_16X16X4_F32
// One wave computes one 16-row x 16-col tile; K=128 in 32 steps of 4.
__global__ __launch_bounds__(256) void gcn_gemm1_wmma(const float* __restrict__ x,
                                                      const float* __restrict__ W1,
                                                      float* __restrict__ h, int n) {
  const int lane = threadIdx.x & 31;
  const int wave = threadIdx.x >> 5;
  const int tile = blockIdx.x * 8 + wave;
  const int tiles = (n + 15) >> 4;
  if (tile >= tiles) return;                      // wave-uniform exit (EXEC stays all-1 for WMMA)

  const int rowBase = tile << 4;
  const int half = lane >> 4;                     // 0: lanes 0-15, 1: lanes 16-31
  const int lmod = lane & 15;
  int arow = rowBase + lmod;
  if (arow >= n) arow = n - 1;                    // clamp loads for tail tile

  v8f acc = {0.f, 0.f, 0.f, 0.f, 0.f, 0.f, 0.f, 0.f};
  const float* xrow = x + (long long)arow * F_DIM;

#pragma unroll
  for (int k = 0; k < F_DIM; k += 4) {
    // A-matrix 16x4 f32: VGPR0 = K(2*half), VGPR1 = K(2*half+1), M = lane%16
    v2f a;
    a.x = xrow[k + half * 2 + 0];
    a.y = xrow[k + half * 2 + 1];
    // B-matrix 4x16 f32: VGPR0 = row K(2*half) col lmod, VGPR1 = row K(2*half+1)
    v2f b;
    const float* bp = W1 + (k + half * 2) * H_DIM + lmod;
    b.x = bp[0];
    b.y = bp[H_DIM];
    acc = __builtin_amdgcn_wmma_f32_16x16x4_f32(false, a, false, b,
                                                (short)0, acc, false, false);
  }

  // D layout: element v -> row rowBase + v + 8*half, col lmod
#pragma unroll
  for (int v = 0; v < 8; ++v) {
    int r = rowBase + v + 8 * half;
    if (r < n) h[(long long)r * H_DIM + lmod] = acc[v];
  }
}

// ---------------------------------------------------------------- layer-1 edge scatter
// thread -> (edge, 4-feature group): 1x b128 gather + 4 hw f32 atomics
__global__ void gcn_scatter1(const int* __restrict__ src, const int* __restrict__ dst,
                             const float* __restrict__ dis, const float* __restrict__ h,
                             float* __restrict__ agg1, int E) {
  int idx = blockIdx.x * blockDim.x + threadIdx.x;
  int e = idx >> 2;
  if (e >= E) return;
  int jg = (idx & 3) * 4;
  int s = src[e], d = dst[e];
  float norm = dis[s] * dis[d];
  const float4 hv = *(const float4*)(h + (long long)s * H_DIM + jg);
  float* ap = agg1 + (long long)d * H_DIM + jg;
  unsafeAtomicAdd(ap + 0, hv.x * norm);
  unsafeAtomicAdd(ap + 1, hv.y * norm);
  unsafeAtomicAdd(ap + 2, hv.z * norm);
  unsafeAtomicAdd(ap + 3, hv.w * norm);
}

// ---------------------------------------------------------------- fused: +self-loop, +b1, ReLU, dot W2
__global__ void gcn_finish1(const float* __restrict__ agg1, const float* __restrict__ h,
                            const float* __restrict__ dis, const float* __restrict__ b1,
                            const float* __restrict__ W2, float* __restrict__ t, int n) {
  int i = blockIdx.x * blockDim.x + threadIdx.x;
  if (i >= n) return;
  float selfn = dis[i] * dis[i];                  // = 1/deg
  const float* ar = agg1 + (long long)i * H_DIM;
  const float* hr = h + (long long)i * H_DIM;
  float acc = 0.0f;
#pragma unroll
  for (int j = 0; j < H_DIM; ++j) {
    float v = ar[j] + hr[j] * selfn + b1[j];
    v = fmaxf(v, 0.0f);                           // ReLU
    acc += v * W2[j];                             // layer-2 GEMM row (H->1)
  }
  t[i] = acc;
}

// ---------------------------------------------------------------- layer-2 edge scatter (scalar)
__global__ void gcn_scatter2(const int* __restrict__ src, const int* __restrict__ dst,
                             const float* __restrict__ dis, const float* __restrict__ t,
                             float* __restrict__ out, int E) {
  int e = blockIdx.x * blockDim.x + threadIdx.x;
  if (e >= E) return;
  int s = src[e], d = dst[e];
  unsafeAtomicAdd(&out[d], t[s] * dis[s] * dis[d]);
}

// ---------------------------------------------------------------- +self-loop term + b2
__global__ void gcn_final(const float* __restrict__ t, const float* __restrict__ dis,
                          const float* __restrict__ b2, float* __restrict__ out, int n) {
  int i = blockIdx.x * blockDim.x + threadIdx.x;
  if (i < n) out[i] += t[i] * dis[i] * dis[i] + b2[0];
}

extern "C" void kernel_launch(void* const* d_in, const int* in_sizes, int n_in,
                              void* d_out, int out_size, void* d_ws, size_t ws_size,
                              hipStream_t stream) {
  const float* x   = (const float*)d_in[0];
  const int*   ei  = (const int*)d_in[1];
  const float* W1  = (const float*)d_in[2];
  const float* b1  = (const float*)d_in[3];
  const float* W2  = (const float*)d_in[4];
  const float* b2  = (const float*)d_in[5];
  float* out = (float*)d_out;

  const int n = in_sizes[0] / F_DIM;      // 100000
  const int E = in_sizes[1] / 2;          // 3200000
  const int* src = ei;                    // edge_index[0]
  const int* dst = ei + E;                // edge_index[1]

  // workspace layout (floats): deg[n] | dis[n] | h[n*16] | agg1[n*16] | t[n]
  float* ws   = (float*)d_ws;
  float* deg  = ws;
  float* dis  = ws + n;
  float* h    = ws + 2 * (size_t)n;
  float* agg1 = ws + 2 * (size_t)n + (size_t)n * H_DIM;
  float* t    = ws + 2 * (size_t)n + 2 * (size_t)n * H_DIM;

  const int B = 256;
  gcn_init<<<(n * H_DIM + B - 1) / B, B, 0, stream>>>(deg, agg1, out, n);
  gcn_degree<<<(E + B - 1) / B, B, 0, stream>>>(dst, deg, E);
  gcn_dis<<<(n + B - 1) / B, B, 0, stream>>>(deg, dis, n);

  const int tiles = (n + 15) / 16;
  gcn_gemm1_wmma<<<(tiles + 7) / 8, B, 0, stream>>>(x, W1, h, n);

  gcn_scatter1<<<((long long)E * 4 + B - 1) / B, B, 0, stream>>>(src, dst, dis, h, agg1, E);
  gcn_finish1<<<(n + B - 1) / B, B, 0, stream>>>(agg1, h, dis, b1, W2, t, n);
  gcn_scatter2<<<(E + B - 1) / B, B, 0, stream>>>(src, dst, dis, t, out, E);
  gcn_final<<<(n + B - 1) / B, B, 0, stream>>>(t, dis, b2, out, n);
}